// update_v_20134806683673
// MI455X (gfx1250) — compile-verified
//
#include <hip/hip_runtime.h>
#include <math.h>

#define N_NODES 50000
#define N_EDGES 600000
#define LN2F    0.6931471805599453f
#define EPSF    1e-5f
#define TS      132   // padded LDS row stride in floats (bank-conflict free, 16B aligned)

typedef __attribute__((ext_vector_type(2))) float v2f;
typedef __attribute__((ext_vector_type(4))) float v4f;
typedef __attribute__((ext_vector_type(8))) float v8f;

// ---------------------------------------------------------------------------
// CSR build: histogram -> block scan -> bucket fill
// ---------------------------------------------------------------------------
__global__ void hist_kernel(const int* __restrict__ dst, int* __restrict__ counts) {
  int g = blockIdx.x * blockDim.x + threadIdx.x;
  if (g < N_EDGES) atomicAdd(&counts[dst[g]], 1);
}

__global__ __launch_bounds__(1024) void scan_kernel(const int* __restrict__ counts,
                                                    int* __restrict__ offsets,
                                                    int* __restrict__ cursor, int n) {
  __shared__ int buf[1024];
  const int tid = threadIdx.x;
  int carry = 0;
  for (int base = 0; base < n; base += 1024) {
    int i = base + tid;
    int x = (i < n) ? counts[i] : 0;
    int val = x;
    buf[tid] = val;
    __syncthreads();
    for (int d = 1; d < 1024; d <<= 1) {
      int y = (tid >= d) ? buf[tid - d] : 0;
      __syncthreads();
      val += y;
      buf[tid] = val;
      __syncthreads();
    }
    if (i < n) { int excl = carry + val - x; offsets[i] = excl; cursor[i] = excl; }
    carry += buf[1023];
    __syncthreads();
  }
  if (tid == 0) offsets[n] = carry;
}

__global__ void fill_kernel(const int* __restrict__ dst, int* __restrict__ cursor,
                            int* __restrict__ edge_ids) {
  int g = blockIdx.x * blockDim.x + threadIdx.x;
  if (g < N_EDGES) {
    int p = atomicAdd(&cursor[dst[g]], 1);
    edge_ids[p] = g;
  }
}

// ---------------------------------------------------------------------------
// fp32 WMMA 16x16 tile GEMM over K=128 (D[m][n] = sum_k A[m][k]*W[n][k])
// A frag: lane lm holds row M=lm, K = k0 + 2*lh + {0,1}  (ds_load_b64)
// B frag: lane lm holds col N=16n+lm, K = k0 + 2*lh + {0,1}
// ---------------------------------------------------------------------------
__device__ __forceinline__ void gemm_tile(const float* __restrict__ tile,
                                          const float* __restrict__ sW,
                                          int lm, int lh, v8f acc[8]) {
#pragma unroll 4
  for (int k0 = 0; k0 < 128; k0 += 4) {
    const v2f aF = *(const v2f*)(tile + lm * TS + k0 + 2 * lh);
#pragma unroll
    for (int n = 0; n < 8; ++n) {
      const v2f bF = *(const v2f*)(sW + (n * 16 + lm) * TS + k0 + 2 * lh);
      acc[n] = __builtin_amdgcn_wmma_f32_16x16x4_f32(
          false, aF, false, bF, (short)0, acc[n], false, false);
    }
  }
}

// ---------------------------------------------------------------------------
// Fused: gather(agg) -> GEMM1 -> softplus -> GEMM2 -> LayerNorm -> v + out
// 256 threads = 8 waves; each wave owns 16 nodes; block owns 128 nodes.
// ---------------------------------------------------------------------------
__global__ __launch_bounds__(256) void fused_kernel(
    const float* __restrict__ v, const float* __restrict__ e,
    const float* __restrict__ W11, const float* __restrict__ b11,
    const float* __restrict__ W12, const float* __restrict__ b12,
    const float* __restrict__ ln_w, const float* __restrict__ ln_b,
    const int* __restrict__ off, const int* __restrict__ edge_ids,
    float* __restrict__ out) {
  extern __shared__ float lds[];
  float* sW    = lds;                    // 128*TS
  float* sTile = sW + 128 * TS;          // 8 * 16*TS
  float* sMu   = sTile + 8 * 16 * TS;    // 128
  float* sRstd = sMu + 128;              // 128
  float* sP    = sRstd + 128;            // 8 * 64 (partial sums)

  const int tid  = threadIdx.x;
  const int w    = tid >> 5;
  const int lane = tid & 31;
  const int lm   = lane & 15;
  const int lh   = lane >> 4;
  float* tile = sTile + w * 16 * TS;
  const int nodeBase = blockIdx.x * 128 + w * 16;

  // stage W11 into LDS (padded rows)
  for (int idx = tid; idx < 128 * 128; idx += 256)
    sW[(idx >> 7) * TS + (idx & 127)] = W11[idx];

  // gather: segment-sum incoming edge rows; lane owns columns [4*lane, 4*lane+4)
  for (int i = 0; i < 16; ++i) {
    int node = nodeBase + i;
    v4f a0 = {0.f, 0.f, 0.f, 0.f};
    v4f a1 = {0.f, 0.f, 0.f, 0.f};
    if (node < N_NODES) {
      int s = off[node], t = off[node + 1];
      int j = s;
      for (; j + 1 < t; j += 2) {
        int e0 = edge_ids[j], e1 = edge_ids[j + 1];
        const v4f x0 = __builtin_nontemporal_load(
            (const v4f*)(e + (size_t)e0 * 128 + lane * 4));
        const v4f x1 = __builtin_nontemporal_load(
            (const v4f*)(e + (size_t)e1 * 128 + lane * 4));
        a0 += x0;
        a1 += x1;
      }
      if (j < t) {
        int e0 = edge_ids[j];
        const v4f x0 = __builtin_nontemporal_load(
            (const v4f*)(e + (size_t)e0 * 128 + lane * 4));
        a0 += x0;
      }
    }
    const v4f a = a0 + a1;
    *(v4f*)(tile + i * TS + lane * 4) = a;
  }
  __syncthreads();

  // ---- layer 1: h1 = shifted_softplus(agg @ W11.T + b11) ----
  const v8f vzero = {0.f, 0.f, 0.f, 0.f, 0.f, 0.f, 0.f, 0.f};
  v8f acc[8];
#pragma unroll
  for (int n = 0; n < 8; ++n) acc[n] = vzero;
  gemm_tile(tile, sW, lm, lh, acc);
  __syncthreads();  // done reading sW and tile

#pragma unroll
  for (int n = 0; n < 8; ++n) {
    float bb = b11[n * 16 + lm];
#pragma unroll
    for (int r = 0; r < 8; ++r) {
      float x  = acc[n][r] + bb;
      float sp = fmaxf(x, 0.f) + log1pf(expf(-fabsf(x))) - LN2F;
      tile[(r + 8 * lh) * TS + n * 16 + lm] = sp;  // D layout: M=r+8*lh, N=16n+lm
    }
  }
  // stage W12 over W11
  for (int idx = tid; idx < 128 * 128; idx += 256)
    sW[(idx >> 7) * TS + (idx & 127)] = W12[idx];
  __syncthreads();

  // ---- layer 2: h2 = h1 @ W12.T + b12 ----
#pragma unroll
  for (int n = 0; n < 8; ++n) acc[n] = vzero;
  gemm_tile(tile, sW, lm, lh, acc);
  __syncthreads();  // done reading tile before overwrite

#pragma unroll
  for (int n = 0; n < 8; ++n) {
    float bb = b12[n * 16 + lm];
#pragma unroll
    for (int r = 0; r < 8; ++r)
      tile[(r + 8 * lh) * TS + n * 16 + lm] = acc[n][r] + bb;
  }
  __syncthreads();

  // ---- LayerNorm: lane reduces half a row (row lm, cols [64*lh, 64*lh+64)) ----
  float s = 0.f, s2 = 0.f;
#pragma unroll
  for (int c = 0; c < 16; ++c) {
    const v4f x4 = *(const v4f*)(tile + lm * TS + lh * 64 + c * 4);
    s  += x4.x + x4.y + x4.z + x4.w;
    s2 += x4.x * x4.x + x4.y * x4.y + x4.z * x4.z + x4.w * x4.w;
  }
  sP[w * 64 + lane * 2]     = s;
  sP[w * 64 + lane * 2 + 1] = s2;
  __syncthreads();
  s  += sP[w * 64 + (lane ^ 16) * 2];
  s2 += sP[w * 64 + (lane ^ 16) * 2 + 1];
  float mu   = s * (1.f / 128.f);
  float var  = s2 * (1.f / 128.f) - mu * mu;
  float rstd = rsqrtf(var + EPSF);
  if (lane < 16) { sMu[w * 16 + lm] = mu; sRstd[w * 16 + lm] = rstd; }
  __syncthreads();

  // ---- residual + affine, coalesced float4 stores ----
  const v4f lw = *(const v4f*)(ln_w + lane * 4);
  const v4f lb = *(const v4f*)(ln_b + lane * 4);
  for (int i = 0; i < 16; ++i) {
    int node = nodeBase + i;
    if (node < N_NODES) {
      float m  = sMu[w * 16 + i];
      float rs = sRstd[w * 16 + i];
      const v4f h4 = *(const v4f*)(tile + i * TS + lane * 4);
      const v4f v4 = *(const v4f*)(v + (size_t)node * 128 + lane * 4);
      v4f o;
      o.x = v4.x + (h4.x - m) * rs * lw.x + lb.x;
      o.y = v4.y + (h4.y - m) * rs * lw.y + lb.y;
      o.z = v4.z + (h4.z - m) * rs * lw.z + lb.z;
      o.w = v4.w + (h4.w - m) * rs * lw.w + lb.w;
      *(v4f*)(out + (size_t)node * 128 + lane * 4) = o;
    }
  }
}

// ---------------------------------------------------------------------------
extern "C" void kernel_launch(void* const* d_in, const int* in_sizes, int n_in,
                              void* d_out, int out_size, void* d_ws, size_t ws_size,
                              hipStream_t stream) {
  const float* v    = (const float*)d_in[0];
  const float* e    = (const float*)d_in[1];
  const int*   eidx = (const int*)d_in[2];   // int32 (jax x64 disabled)
  // d_in[3] = v1_size (unused: same weights applied to both halves)
  const float* W11  = (const float*)d_in[4];
  const float* b11  = (const float*)d_in[5];
  const float* W12  = (const float*)d_in[6];
  const float* b12  = (const float*)d_in[7];
  const float* lnw  = (const float*)d_in[8];
  const float* lnb  = (const float*)d_in[9];
  float* out = (float*)d_out;
  const int* dst = eidx + N_EDGES;           // edge_index[1]

  // workspace layout (~3 MB): counts | offsets | cursor | edge_ids
  int* counts   = (int*)d_ws;                // 50001 (padded to 50016)
  int* offsets  = counts + 50016;            // 50001
  int* cursor   = offsets + 50016;           // 50000
  int* edge_ids = cursor + 50016;            // 600000

  (void)hipMemsetAsync(counts, 0, sizeof(int) * 50001, stream);
  hist_kernel<<<(N_EDGES + 255) / 256, 256, 0, stream>>>(dst, counts);
  scan_kernel<<<1, 1024, 0, stream>>>(counts, offsets, cursor, N_NODES);
  fill_kernel<<<(N_EDGES + 255) / 256, 256, 0, stream>>>(dst, cursor, edge_ids);

  size_t ldsBytes = (size_t)(128 * TS + 8 * 16 * TS + 128 + 128 + 8 * 64) * sizeof(float);
  fused_kernel<<<(N_NODES + 127) / 128, 256, ldsBytes, stream>>>(
      v, e, W11, b11, W12, b12, lnw, lnb, offsets, edge_ids, out);
}